// VectorQuantizerEMA_67147518706259
// MI455X (gfx1250) — compile-verified
//
#include <hip/hip_runtime.h>
#include <hip/hip_bf16.h>

// ---- problem constants --------------------------------------------------
#define NUM_EMB   8192
#define EMB_DIM   256
#define N_ROWS    16384      // 16 * 1024
#define TOPK      3
#define Q_ELEMS   (N_ROWS * TOPK * EMB_DIM)   // 12,582,912

typedef __attribute__((ext_vector_type(2))) float v2f;
typedef __attribute__((ext_vector_type(4))) float v4f;
typedef __attribute__((ext_vector_type(8))) float v8f;

// ---- 1) zero scratch (counts + loss accumulator) ------------------------
__global__ void vq_init(float* __restrict__ counts, float* __restrict__ accum) {
    int i = blockIdx.x * blockDim.x + threadIdx.x;
    if (i < NUM_EMB) counts[i] = 0.0f;
    if (i == 0) accum[0] = 0.0f;
}

// ---- 2) row squared norms: x2[16384], e2[8192] (one wave per row) -------
__global__ __launch_bounds__(256)
void vq_norms(const float* __restrict__ X, const float* __restrict__ E,
              float* __restrict__ x2, float* __restrict__ e2) {
    const int wave = threadIdx.x >> 5;
    const int lane = threadIdx.x & 31;
    const int r    = blockIdx.x * 8 + wave;          // 0 .. 24575
    const float* src = (r < N_ROWS) ? (X + (size_t)r * EMB_DIM)
                                    : (E + (size_t)(r - N_ROWS) * EMB_DIM);
    float s = 0.0f;
    #pragma unroll
    for (int i = 0; i < EMB_DIM; i += 32) {
        float v = src[i + lane];
        s += v * v;
    }
    #pragma unroll
    for (int m = 16; m >= 1; m >>= 1) s += __shfl_xor(s, m, 32);
    if (lane == 0) {
        if (r < N_ROWS) x2[r] = s;
        else            e2[r - N_ROWS] = s;
    }
}

// ---- 3) distance GEMM via V_WMMA_F32_16X16X4_F32 ------------------------
// dist[m][n] = x2[m] + e2[n] - 2 * dot(X[m,:], E[n,:])
// Block tile: 64 (M) x 128 (N); 8 waves, each wave = 2x2 grid of 16x16 tiles.
#define MT   64
#define NT   128
#define KC   64
#define LDSS 68   // padded row stride (floats): 68*4 bytes keeps b64 aligned,
                  // 4*row mod 64 spreads banks across the 16 rows per fetch

__global__ __launch_bounds__(256)
void vq_dist_wmma(const float* __restrict__ X, const float* __restrict__ E,
                  const float* __restrict__ x2, const float* __restrict__ e2,
                  float* __restrict__ dist) {
    __shared__ float Xs[MT * LDSS];   // 17,408 B
    __shared__ float Es[NT * LDSS];   // 34,816 B

    const int tid   = threadIdx.x;
    const int wave  = tid >> 5;
    const int lane  = tid & 31;
    const int wm    = wave >> 2;          // 0..1  (which 32 rows)
    const int wn    = wave & 3;           // 0..3  (which 32 cols)
    const int row16 = lane & 15;
    const int kh    = lane >> 4;          // K half-select per ISA A/B layout

    const int m0 = blockIdx.y * MT;
    const int n0 = blockIdx.x * NT;

    // staging-load coordinates (float4 per thread per row-group)
    const int lr = tid >> 4;              // 0..15
    const int lc = (tid & 15) * 4;        // 0..60

    v8f acc[2][2] = {};

    for (int kc = 0; kc < EMB_DIM; kc += KC) {
        #pragma unroll
        for (int i = 0; i < MT; i += 16) {
            const int r = lr + i;
            *(v4f*)&Xs[r * LDSS + lc] =
                *(const v4f*)&X[(size_t)(m0 + r) * EMB_DIM + kc + lc];
        }
        #pragma unroll
        for (int i = 0; i < NT; i += 16) {
            const int r = lr + i;
            *(v4f*)&Es[r * LDSS + lc] =
                *(const v4f*)&E[(size_t)(n0 + r) * EMB_DIM + kc + lc];
        }
        __syncthreads();

        #pragma unroll
        for (int kb = 0; kb < KC; kb += 4) {
            const int ac = kb + 2 * kh;   // ISA 16x4 layout: v0->K{0,2}, v1->K{1,3}
            v2f a0 = *(const v2f*)&Xs[(wm * 32 +      row16) * LDSS + ac];
            v2f a1 = *(const v2f*)&Xs[(wm * 32 + 16 + row16) * LDSS + ac];
            v2f b0 = *(const v2f*)&Es[(wn * 32 +      row16) * LDSS + ac];
            v2f b1 = *(const v2f*)&Es[(wn * 32 + 16 + row16) * LDSS + ac];
            acc[0][0] = __builtin_amdgcn_wmma_f32_16x16x4_f32(
                false, a0, false, b0, (short)0, acc[0][0], false, false);
            acc[0][1] = __builtin_amdgcn_wmma_f32_16x16x4_f32(
                false, a0, false, b1, (short)0, acc[0][1], false, false);
            acc[1][0] = __builtin_amdgcn_wmma_f32_16x16x4_f32(
                false, a1, false, b0, (short)0, acc[1][0], false, false);
            acc[1][1] = __builtin_amdgcn_wmma_f32_16x16x4_f32(
                false, a1, false, b1, (short)0, acc[1][1], false, false);
        }
        __syncthreads();
    }

    // Epilogue: C/D layout (v8f): VGPR r, lanes0-15 -> M=r, lanes16-31 -> M=r+8
    #pragma unroll
    for (int i = 0; i < 2; ++i) {
        #pragma unroll
        for (int j = 0; j < 2; ++j) {
            const int gn  = n0 + wn * 32 + j * 16 + row16;
            const float en = e2[gn];
            #pragma unroll
            for (int r = 0; r < 8; ++r) {
                const int gm = m0 + wm * 32 + i * 16 + r + 8 * kh;
                dist[(size_t)gm * NUM_EMB + gn] = x2[gm] + en - 2.0f * acc[i][j][r];
            }
        }
    }
}

// ---- 4) stable top-3 (smallest distance, lowest index on ties) ----------
__global__ __launch_bounds__(256)
void vq_top3(const float* __restrict__ dist, float* __restrict__ out_idx_f,
             int* __restrict__ ws_idx) {
    __shared__ float sd[256 * 3];
    __shared__ int   si[256 * 3];
    const int n = blockIdx.x;
    const float* row = dist + (size_t)n * NUM_EMB;

    float d0 = 3.4e38f, d1 = 3.4e38f, d2 = 3.4e38f;
    int   i0 = -1,      i1 = -1,      i2 = -1;
    for (int k = threadIdx.x; k < NUM_EMB; k += 256) {
        const float d = row[k];
        if (d < d0)      { d2 = d1; i2 = i1; d1 = d0; i1 = i0; d0 = d; i0 = k; }
        else if (d < d1) { d2 = d1; i2 = i1; d1 = d;  i1 = k; }
        else if (d < d2) { d2 = d;  i2 = k; }
    }
    sd[threadIdx.x * 3 + 0] = d0; si[threadIdx.x * 3 + 0] = i0;
    sd[threadIdx.x * 3 + 1] = d1; si[threadIdx.x * 3 + 1] = i1;
    sd[threadIdx.x * 3 + 2] = d2; si[threadIdx.x * 3 + 2] = i2;
    __syncthreads();

    if (threadIdx.x == 0) {
        float b0 = 3.4e38f, b1 = 3.4e38f, b2 = 3.4e38f;
        int   j0 = 0x7fffffff, j1 = 0x7fffffff, j2 = 0x7fffffff;
        for (int t = 0; t < 256 * 3; ++t) {
            const float d = sd[t];
            const int   k = si[t];
            if (k < 0) continue;
            if (d < b0 || (d == b0 && k < j0)) {
                b2 = b1; j2 = j1; b1 = b0; j1 = j0; b0 = d; j0 = k;
            } else if (d < b1 || (d == b1 && k < j1)) {
                b2 = b1; j2 = j1; b1 = d; j1 = k;
            } else if (d < b2 || (d == b2 && k < j2)) {
                b2 = d; j2 = k;
            }
        }
        out_idx_f[n * 3 + 0] = (float)j0;  ws_idx[n * 3 + 0] = j0;
        out_idx_f[n * 3 + 1] = (float)j1;  ws_idx[n * 3 + 1] = j1;
        out_idx_f[n * 3 + 2] = (float)j2;  ws_idx[n * 3 + 2] = j2;
    }
}

// ---- 5) gather quantized_st, loss partials, histogram -------------------
// quantized_st value == quantized (straight-through): out = E[idx]
__global__ __launch_bounds__(256)
void vq_quantize(const float* __restrict__ X, const float* __restrict__ E,
                 const int* __restrict__ idx, float* __restrict__ outq,
                 float* __restrict__ counts, float* __restrict__ accum) {
    __shared__ float red[8];
    const int slot = blockIdx.x;          // 0 .. N_ROWS*TOPK-1
    const int n    = slot / TOPK;
    const int e    = idx[slot];
    const int c    = threadIdx.x;

    const float q = E[(size_t)e * EMB_DIM + c];
    const float x = X[(size_t)n * EMB_DIM + c];
    outq[(size_t)slot * EMB_DIM + c] = q;

    float p = (q - x) * (q - x);
    #pragma unroll
    for (int m = 16; m >= 1; m >>= 1) p += __shfl_xor(p, m, 32);
    const int lane = threadIdx.x & 31, wave = threadIdx.x >> 5;
    if (lane == 0) red[wave] = p;
    __syncthreads();
    if (threadIdx.x == 0) {
        float s = 0.0f;
        #pragma unroll
        for (int w = 0; w < 8; ++w) s += red[w];
        atomicAdd(accum, s);
        atomicAdd(&counts[e], 1.0f);
    }
}

// ---- 6) loss + perplexity ----------------------------------------------
__global__ __launch_bounds__(256)
void vq_finalize(const float* __restrict__ counts, const float* __restrict__ accum,
                 float* __restrict__ out_loss, float* __restrict__ out_perp) {
    __shared__ float red[8];
    float h = 0.0f;
    for (int k = threadIdx.x; k < NUM_EMB; k += 256) {
        const float p = counts[k] * (1.0f / (float)N_ROWS);
        h -= p * logf(p + 1e-10f);
    }
    #pragma unroll
    for (int m = 16; m >= 1; m >>= 1) h += __shfl_xor(h, m, 32);
    const int lane = threadIdx.x & 31, wave = threadIdx.x >> 5;
    if (lane == 0) red[wave] = h;
    __syncthreads();
    if (threadIdx.x == 0) {
        float s = 0.0f;
        #pragma unroll
        for (int w = 0; w < 8; ++w) s += red[w];
        out_perp[0] = expf(s);
        out_loss[0] = 0.25f * accum[0] / (float)Q_ELEMS;
    }
}

// ---- launch -------------------------------------------------------------
extern "C" void kernel_launch(void* const* d_in, const int* in_sizes, int n_in,
                              void* d_out, int out_size, void* d_ws, size_t ws_size,
                              hipStream_t stream) {
    const float* X = (const float*)d_in[0];   // (16,1024,256)
    const float* E = (const float*)d_in[1];   // (8192,256)

    float* out      = (float*)d_out;
    float* out_loss = out;                              // [1]
    float* out_q    = out + 1;                          // [12,582,912]
    float* out_perp = out + 1 + Q_ELEMS;                // [1]
    float* out_idx  = out_perp + 1;                     // [49,152] (as float)
    float* out_dist = out_idx + N_ROWS * TOPK;          // [134,217,728]

    float* ws        = (float*)d_ws;
    float* ws_x2     = ws;                              // 16384
    float* ws_e2     = ws + N_ROWS;                     // 8192
    float* ws_counts = ws + N_ROWS + NUM_EMB;           // 8192
    float* ws_accum  = ws + N_ROWS + 2 * NUM_EMB;       // 1 (+3 pad)
    int*   ws_idx    = (int*)(ws + N_ROWS + 2 * NUM_EMB + 4);  // 49152 ints

    vq_init<<<(NUM_EMB + 255) / 256, 256, 0, stream>>>(ws_counts, ws_accum);
    vq_norms<<<(N_ROWS + NUM_EMB) / 8, 256, 0, stream>>>(X, E, ws_x2, ws_e2);

    dim3 grid(NUM_EMB / NT, N_ROWS / MT);   // (64, 256)
    vq_dist_wmma<<<grid, 256, 0, stream>>>(X, E, ws_x2, ws_e2, out_dist);

    vq_top3<<<N_ROWS, 256, 0, stream>>>(out_dist, out_idx, ws_idx);
    vq_quantize<<<N_ROWS * TOPK, 256, 0, stream>>>(X, E, ws_idx, out_q,
                                                   ws_counts, ws_accum);
    vq_finalize<<<1, 256, 0, stream>>>(ws_counts, ws_accum, out_loss, out_perp);
}